// GreyBox_21320217657990
// MI455X (gfx1250) — compile-verified
//
#include <hip/hip_runtime.h>
#include <hip/hip_bf16.h>

#define SDIM   100     // state dim
#define SPAD   112     // 7*16 (row padding for 16-row WMMA tiles)
#define HID    8
#define NOBS   12
#define DTC    0.01f
#define NCHUNK 25      // SDIM / 4 — K needs NO padding (100 = 25 * 4)
#define NWA    7       // waves carrying A row-tiles; wave 7 carries W1

typedef float v2f __attribute__((ext_vector_type(2)));
typedef float v8f __attribute__((ext_vector_type(8)));

__global__ __launch_bounds__(256, 1)
void greybox_scan_kernel(const float* __restrict__ c0,
                         const float* __restrict__ w,
                         const float* __restrict__ u,
                         const float* __restrict__ A,
                         const float* __restrict__ Bw,
                         const float* __restrict__ W1,
                         const float* __restrict__ b1,
                         const float* __restrict__ W2,
                         const float* __restrict__ b2,
                         const int*   __restrict__ obs,
                         float* __restrict__ out_c,
                         float* __restrict__ out_y,
                         int T)
{
    __shared__ float cS[SDIM];          // current state
    __shared__ float zS[SDIM];          // all-zero region: B-operand source for masked lanes
    __shared__ float hS[HID];           // tanh hidden
    __shared__ float accS[SPAD];        // A@c + dt*W2@h column (row-padded)
    __shared__ float uS[2][SDIM];       // double-buffered u_t prefetch
    __shared__ float wS[2][2];          // double-buffered w_t prefetch
    __shared__ float BwS[2 * SDIM];
    __shared__ float b2S[SDIM];         // dt * b2
    __shared__ float b1S[HID];
    __shared__ int   obsS[NOBS];

    const int tid  = threadIdx.x;
    const int wave = tid >> 5;
    const int lane = tid & 31;
    const int hi   = (lane >> 4) & 1;   // 0: lanes 0-15 (K pair 0,1), 1: lanes 16-31 (K pair 2,3)
    const int rloc = lane & 15;         // row within 16-row tile

    // ---------------- one-time setup ----------------
    for (int i = tid; i < 2 * SDIM; i += 256) BwS[i] = Bw[i];
    for (int i = tid; i < SDIM;     i += 256) { b2S[i] = DTC * b2[i]; zS[i] = 0.0f; cS[i] = c0[i]; }
    for (int i = tid; i < HID;      i += 256) b1S[i] = b1[i];
    for (int i = tid; i < NOBS;     i += 256) obsS[i] = obs[i];
    for (int i = tid; i < SPAD;     i += 256) accS[i] = 0.0f;
    if (tid < SDIM) { uS[0][tid] = u[tid]; uS[1][tid] = 0.0f; }
    if (tid < 2)    { wS[0][tid] = w[tid]; wS[1][tid] = 0.0f; }

    // Per-lane B-operand base pointers: lanes 0/16 read the live vector,
    // every other lane reads from the zero region (no per-chunk masking ALU).
    const float* bBaseC = (rloc == 0) ? (cS + 2 * hi) : zS;
    const float* bBaseH = (rloc == 0) ? (hS + 2 * hi) : zS;

    // ---- resident weight fragments (16x4 f32 A-operand layout) ----
    // lane L<16 : rows M=L,      VGPR0=K0, VGPR1=K1
    // lane L>=16: rows M=L-16,   VGPR0=K2, VGPR1=K3
    v2f afrag[NCHUNK] = {};   // waves 0..6: A tile; wave 7: W1 (rows padded to 16)
    v2f w2frag[2]     = {};   // waves 0..6: dt * W2 tile (K = hidden, 2 chunks)

    if (wave < NWA) {
        const int r = wave * 16 + rloc;
        #pragma unroll
        for (int j = 0; j < NCHUNK; ++j) {
            const int k = 4 * j + 2 * hi;      // always < SDIM (no K padding)
            v2f a; a.x = 0.0f; a.y = 0.0f;
            if (r < SDIM) {
                a.x = A[r * SDIM + k];
                a.y = A[r * SDIM + k + 1];
            }
            afrag[j] = a;
        }
        #pragma unroll
        for (int jj = 0; jj < 2; ++jj) {
            const int k = 4 * jj + 2 * hi;      // 0..7 < HID always
            v2f a; a.x = 0.0f; a.y = 0.0f;
            if (r < SDIM) {
                a.x = DTC * W2[r * HID + k];
                a.y = DTC * W2[r * HID + k + 1];
            }
            w2frag[jj] = a;
        }
    } else {
        const int rw = rloc;                    // W1 row, valid if < HID
        #pragma unroll
        for (int j = 0; j < NCHUNK; ++j) {
            const int k = 4 * j + 2 * hi;
            v2f a; a.x = 0.0f; a.y = 0.0f;
            if (rw < HID) {
                a.x = W1[rw * SDIM + k];
                a.y = W1[rw * SDIM + k + 1];
            }
            afrag[j] = a;
        }
    }
    __syncthreads();

    // ---------------- sequential scan ----------------
    for (int t = 0; t < T; ++t) {
        const int buf = t & 1;

        // prefetch u[t+1], w[t+1] into the other buffer; these global loads
        // retire in the shadow of this step's WMMA chains
        if (t + 1 < T) {
            if (tid < SDIM)
                uS[buf ^ 1][tid] = u[(size_t)(t + 1) * SDIM + tid];
            else if (tid < SDIM + 2)
                wS[buf ^ 1][tid - SDIM] = w[(size_t)(t + 1) * 2 + (tid - SDIM)];
        }

        // 4 independent WMMA accumulate chains (depth <= 7) to keep the XDL
        // pipe issue-bound rather than dependent-latency-bound.
        v8f acc[4];
        #pragma unroll
        for (int q = 0; q < 4; ++q)
            acc[q] = (v8f){0.f, 0.f, 0.f, 0.f, 0.f, 0.f, 0.f, 0.f};

        if (wave == NWA) {
            // wave 7: emit y_{t-1} (cS still holds c_{t-1}), then h = tanh(W1 c + b1)
            if (t > 0 && lane < NOBS)
                out_y[(size_t)(t - 1) * NOBS + lane] = cS[obsS[lane]];
            #pragma unroll
            for (int j = 0; j < NCHUNK; ++j) {
                const v2f b = *(const v2f*)(bBaseC + 4 * j);
                acc[j & 3] = __builtin_amdgcn_wmma_f32_16x16x4_f32(
                                 false, afrag[j], false, b, (short)0, acc[j & 3],
                                 false, false);
            }
            const v8f s = (acc[0] + acc[1]) + (acc[2] + acc[3]);
            if (lane == 0) {                 // D col 0, M=0..7 -> all of HID
                #pragma unroll
                for (int i = 0; i < HID; ++i)
                    hS[i] = tanhf(s[i] + b1S[i]);
            }
        } else {
            // waves 0..6: acc = A_tile @ c
            #pragma unroll
            for (int j = 0; j < NCHUNK; ++j) {
                const v2f b = *(const v2f*)(bBaseC + 4 * j);
                acc[j & 3] = __builtin_amdgcn_wmma_f32_16x16x4_f32(
                                 false, afrag[j], false, b, (short)0, acc[j & 3],
                                 false, false);
            }
        }
        __syncthreads();   // hS ready

        if (wave < NWA) {
            v8f s = (acc[0] + acc[1]) + (acc[2] + acc[3]);
            // fold in dt * W2 @ h (2 chunks), then spill D column 0 to LDS
            #pragma unroll
            for (int jj = 0; jj < 2; ++jj) {
                const v2f b = *(const v2f*)(bBaseH + 4 * jj);
                s = __builtin_amdgcn_wmma_f32_16x16x4_f32(
                        false, w2frag[jj], false, b, (short)0, s, false, false);
            }
            if (rloc == 0) {                 // lanes 0 (M 0..7) and 16 (M 8..15)
                const int base = wave * 16 + hi * 8;
                #pragma unroll
                for (int i = 0; i < 8; ++i)
                    accS[base + i] = s[i];
            }
        }
        __syncthreads();   // accS ready

        // elementwise finish: c_t = acc + Bw*w_t + u_t + dt*b2 ; store state + output
        if (tid < SDIM) {
            const float cn = accS[tid]
                           + BwS[2 * tid]     * wS[buf][0]
                           + BwS[2 * tid + 1] * wS[buf][1]
                           + uS[buf][tid]
                           + b2S[tid];
            cS[tid] = cn;
            out_c[(size_t)t * SDIM + tid] = cn;
        }
        __syncthreads();   // cS ready for next step
    }

    // final observation row
    if (tid < NOBS)
        out_y[(size_t)(T - 1) * NOBS + tid] = cS[obsS[tid]];
}

extern "C" void kernel_launch(void* const* d_in, const int* in_sizes, int n_in,
                              void* d_out, int out_size, void* d_ws, size_t ws_size,
                              hipStream_t stream) {
    const float* c0 = (const float*)d_in[0];
    const float* w  = (const float*)d_in[1];
    const float* u  = (const float*)d_in[2];
    const float* A  = (const float*)d_in[3];
    const float* Bw = (const float*)d_in[4];
    const float* W1 = (const float*)d_in[5];
    const float* b1 = (const float*)d_in[6];
    const float* W2 = (const float*)d_in[7];
    const float* b2 = (const float*)d_in[8];
    const int*  obs = (const int*)  d_in[9];

    const int T = in_sizes[1] / 2;           // w is (T, 2)
    float* out_c = (float*)d_out;            // (T, 100) flat
    float* out_y = out_c + (size_t)T * SDIM; // (T, 12) flat

    greybox_scan_kernel<<<1, 256, 0, stream>>>(c0, w, u, A, Bw, W1, b1, W2, b2,
                                               obs, out_c, out_y, T);
}